// SpectralConv_48928267436259
// MI455X (gfx1250) — compile-verified
//
#include <hip/hip_runtime.h>

// ---------------------------------------------------------------------------
// Spectral graph conv: relu( U @ (g[:,None] * (U^T @ x)) )
//   U: [50000, 512] f32, g: [512] f32, x: [50000, 128] f32 -> out [50000,128]
// fp32 WMMA (v_wmma_f32_16x16x4_f32); async global->LDS double buffering.
// Workspace: 66*512*128*4 (partials) + 512*128*4 (scaled specT) ~= 16.8 MB
// ---------------------------------------------------------------------------

typedef __attribute__((ext_vector_type(2))) float v2f;
typedef __attribute__((ext_vector_type(8))) float v8f;
typedef __attribute__((__vector_size__(16))) int v4i;

#define N_NODES 50000
#define K_EIG   512
#define F_DIM   128

#define NCHUNK  66      // split-K chunks over the node dimension
#define CH_ROWS 768     // rows per chunk (66*768 = 50688 >= 50000)
#define RB      64      // row-block staged in LDS per pipeline step (stage 1)
#define KG      4       // K groups (4 * 128 = 512)
#define KW      128     // K columns per group

#define S1      132     // stage-1 LDS row stride (128 + pad, mult of 4)
#define TILE1   (RB * S1)

#define KCH     32      // stage-2 K chunk per pipeline step
#define S2U     36      // stage-2 U-tile LDS row stride (32 + pad, mult of 4)
#define TILE2   (128 * S2U)
#define SPEC_S  516     // stage-2 spec LDS stride (512 + pad, mult of 4)

#if defined(__AMDGCN__) && __has_builtin(__builtin_amdgcn_global_load_async_to_lds_b128)
#define HAVE_ASYNC_LDS 1
#endif

#define AS1 __attribute__((address_space(1)))
#define AS3 __attribute__((address_space(3)))

// 16-byte global -> LDS copy. Async (ASYNCcnt-tracked, no VGPR round trip)
// when the gfx1250 builtin is available; plain copy otherwise.
__device__ __forceinline__ void copy16(const float* g, float* l) {
#if defined(HAVE_ASYNC_LDS)
  __builtin_amdgcn_global_load_async_to_lds_b128(
      (AS1 v4i*)g, (AS3 v4i*)l, 0, 0);
#else
  *reinterpret_cast<float4*>(l) = *reinterpret_cast<const float4*>(g);
#endif
}

__device__ __forceinline__ void zero16(float* l) {
  float4 z = {0.f, 0.f, 0.f, 0.f};
  *reinterpret_cast<float4*>(l) = z;
}

__device__ __forceinline__ void wait_async() {
#if defined(HAVE_ASYNC_LDS)
# if __has_builtin(__builtin_amdgcn_s_wait_asynccnt)
  __builtin_amdgcn_s_wait_asynccnt(0);
# else
  asm volatile("s_wait_asynccnt 0x0" ::: "memory");
# endif
#endif
}

// ---------------------------------------------------------------------------
// Stage 1: partial[c, k, f] = sum_{r in chunk c} U[r, k] * x[r, f]
// grid (NCHUNK, KG), 256 threads = 8 waves; wave w owns k-strip [16w, 16w+16)
// of this block's 128-wide K group, full F=128 (8 accumulator tiles).
// LDS: 2 x (U row-block [64][S1] + x row-block [64][S1]), double buffered.
// ---------------------------------------------------------------------------
__device__ __forceinline__ void s1_fill(const float* __restrict__ U,
                                        const float* __restrict__ X,
                                        int row0, int kbase,
                                        float* ut, float* xt, int tid) {
  // KW == F_DIM == 128: one index space covers both tiles. 64*128/4 = 2048.
  for (int i = tid; i < (KW * RB) / 4; i += 256) {
    const int idx = i * 4;
    const int r   = idx >> 7;
    const int kl  = idx & (KW - 1);
    const int gr  = row0 + r;
    float* du = &ut[r * S1 + kl];
    float* dx = &xt[r * S1 + kl];
    if (gr < N_NODES) {
      copy16(&U[(size_t)gr * K_EIG + kbase + kl], du);
      copy16(&X[(size_t)gr * F_DIM + kl], dx);
    } else {
      zero16(du);
      zero16(dx);
    }
  }
}

extern "C" __global__ void __launch_bounds__(256)
spec_stage1(const float* __restrict__ U, const float* __restrict__ X,
            float* __restrict__ partials) {
  extern __shared__ float lds[];
  // buffer b: ut = lds + b*2*TILE1, xt = ut + TILE1

  const int tid  = threadIdx.x;
  const int lane = tid & 31;
  const int wave = tid >> 5;
  const int lo   = lane & 15;
  const int hi   = lane >> 4;

  const int chunk = blockIdx.x;
  const int kg    = blockIdx.y;
  const int kbase = kg * KW;
  const int row0  = chunk * CH_ROWS;
  const int kk    = wave * 16;

  v8f acc[8] = {};

  // prologue: fill buffer 0
  s1_fill(U, X, row0, kbase, lds, lds + TILE1, tid);
  wait_async();
  __syncthreads();

  for (int rb = 0; rb < CH_ROWS; rb += RB) {
    const int cur = (rb / RB) & 1;
    float* ut = lds + cur * 2 * TILE1;
    float* xt = ut + TILE1;
    // issue async fill of the next row-block into the other buffer
    if (rb + RB < CH_ROWS) {
      float* nut = lds + (cur ^ 1) * 2 * TILE1;
      s1_fill(U, X, row0 + rb + RB, kbase, nut, nut + TILE1, tid);
    }

    #pragma unroll 4
    for (int rr = 0; rr < RB; rr += 4) {
      const int rbase = (rr + 2 * hi) * S1;
      // A (16x4): A[m][j] = U[row rr+j][k kk+m]; lane: m=lo, j=2*hi(+1)
      const float a0 = ut[rbase + kk + lo];
      const float a1 = ut[rbase + S1 + kk + lo];
      const v2f A = {a0, a1};
      #pragma unroll
      for (int f = 0; f < 8; ++f) {
        // B (4x16): B[j][n] = x[row rr+j][f*16+n]; lane: n=lo, j=2*hi(+1)
        const float b0 = xt[rbase + f * 16 + lo];
        const float b1 = xt[rbase + S1 + f * 16 + lo];
        const v2f B = {b0, b1};
        acc[f] = __builtin_amdgcn_wmma_f32_16x16x4_f32(
            false, A, false, B, (short)0, acc[f], false, false);
      }
    }
    wait_async();
    __syncthreads();
  }

  // write partial tile: each (c,k,f) owned by exactly one thread
  float* p = partials + ((size_t)chunk * K_EIG + kbase + kk) * F_DIM;
  #pragma unroll
  for (int f = 0; f < 8; ++f) {
    #pragma unroll
    for (int v = 0; v < 8; ++v) {
      const int m = v + 8 * hi;  // C/D layout: row = vgpr + 8*(lane>=16)
      p[(size_t)m * F_DIM + f * 16 + lo] = acc[f][v];
    }
  }
}

// ---------------------------------------------------------------------------
// Stage 1b: specT[f, k] = g[k] * sum_c partial[c, k, f]  (fixed order -> det.)
// ---------------------------------------------------------------------------
extern "C" __global__ void __launch_bounds__(256)
spec_reduce(const float* __restrict__ partials, const float* __restrict__ g,
            float* __restrict__ specT) {
  const int idx = blockIdx.x * 256 + threadIdx.x;  // 0 .. 512*128-1
  const int k = idx >> 7;
  const int f = idx & (F_DIM - 1);
  float s = 0.f;
  for (int c = 0; c < NCHUNK; ++c)
    s += partials[((size_t)c * K_EIG + k) * F_DIM + f];
  specT[f * K_EIG + k] = s * g[k];
}

// ---------------------------------------------------------------------------
// Stage 2: out = relu(U @ spec_scaled). Grid: ceil(N/128) blocks, 8 waves.
// LDS: scaled spectrum (f-major, resident, 258 KB) + 2 x U tile [128][S2U]
// (async double buffered). ~301 KB total (CDNA5 WGP: 320 KB).
// ---------------------------------------------------------------------------
__device__ __forceinline__ void s2_fill(const float* __restrict__ U,
                                        int row0, int kc0, float* uL, int tid) {
  for (int i = tid; i < (128 * KCH) / 4; i += 256) {
    const int idx = i * 4;
    const int r   = idx >> 5;       // idx / KCH
    const int kl  = idx & (KCH - 1);
    const int gr  = row0 + r;
    float* d = &uL[r * S2U + kl];
    if (gr < N_NODES)
      copy16(&U[(size_t)gr * K_EIG + kc0 + kl], d);
    else
      zero16(d);
  }
}

extern "C" __global__ void __launch_bounds__(256)
spec_stage2(const float* __restrict__ U, const float* __restrict__ specT,
            float* __restrict__ out) {
  extern __shared__ float lds[];
  float* specL = lds;                    // [F_DIM][SPEC_S]
  float* uB0   = lds + F_DIM * SPEC_S;   // [128][S2U] buffer 0
  float* uB1   = uB0 + TILE2;            // [128][S2U] buffer 1

  const int tid  = threadIdx.x;
  const int lane = tid & 31;
  const int wave = tid >> 5;
  const int lo   = lane & 15;
  const int hi   = lane >> 4;
  const int row0 = blockIdx.x * 128;

  // prologue: async-fill U chunk 0, then (overlapped) stage spec into LDS
  s2_fill(U, row0, 0, uB0, tid);
  for (int i = tid; i < (F_DIM * K_EIG) / 4; i += 256) {
    const int idx = i * 4;
    const int f = idx >> 9;          // idx / K_EIG
    const int k = idx & (K_EIG - 1);
    const float4 v = *reinterpret_cast<const float4*>(&specT[f * K_EIG + k]);
    *reinterpret_cast<float4*>(&specL[f * SPEC_S + k]) = v;
  }
  wait_async();
  __syncthreads();

  v8f acc[8] = {};

  for (int ci = 0; ci < K_EIG / KCH; ++ci) {
    float* cur = (ci & 1) ? uB1 : uB0;
    if (ci + 1 < K_EIG / KCH)
      s2_fill(U, row0, (ci + 1) * KCH, (ci & 1) ? uB0 : uB1, tid);

    const int kg0 = ci * KCH;
    #pragma unroll
    for (int kc = 0; kc < KCH; kc += 4) {
      // A (16x4): A[m][j] = U[row0+16w+m][kg0+kc+j]; lane: m=lo, j=2*hi(+1)
      const float2 a = *reinterpret_cast<const float2*>(
          &cur[(wave * 16 + lo) * S2U + kc + 2 * hi]);
      const v2f A = {a.x, a.y};
      #pragma unroll
      for (int f = 0; f < 8; ++f) {
        // B (4x16): B[j][n] = spec[kg0+kc+j][f*16+n]; lane: n=lo, j=2*hi(+1)
        const float2 b = *reinterpret_cast<const float2*>(
            &specL[(f * 16 + lo) * SPEC_S + kg0 + kc + 2 * hi]);
        const v2f B = {b.x, b.y};
        acc[f] = __builtin_amdgcn_wmma_f32_16x16x4_f32(
            false, A, false, B, (short)0, acc[f], false, false);
      }
    }
    wait_async();
    __syncthreads();
  }

  // ReLU epilogue + store
  #pragma unroll
  for (int f = 0; f < 8; ++f) {
    #pragma unroll
    for (int v = 0; v < 8; ++v) {
      const int m  = v + 8 * hi;
      const int gr = row0 + wave * 16 + m;
      if (gr < N_NODES) {
        const float val = acc[f][v];
        out[(size_t)gr * F_DIM + f * 16 + lo] = val > 0.f ? val : 0.f;
      }
    }
  }
}

// ---------------------------------------------------------------------------
extern "C" void kernel_launch(void* const* d_in, const int* in_sizes, int n_in,
                              void* d_out, int out_size, void* d_ws, size_t ws_size,
                              hipStream_t stream) {
  (void)in_sizes; (void)n_in; (void)out_size; (void)ws_size;
  const float* U = (const float*)d_in[0];   // eigenvectors [N, K]
  const float* g = (const float*)d_in[1];   // spectral_filters [K]
  const float* X = (const float*)d_in[2];   // x [N, F]
  float* out = (float*)d_out;

  float* partials = (float*)d_ws;                               // [NCHUNK,K,F]
  float* specT    = partials + (size_t)NCHUNK * K_EIG * F_DIM;  // [F,K]

  const size_t sh1 = (size_t)4 * TILE1 * sizeof(float);                    // 135,168 B
  const size_t sh2 = (size_t)(F_DIM * SPEC_S + 2 * TILE2) * sizeof(float); // 301,056 B

  // Allow >64KB dynamic LDS (CDNA5 WGP: 320 KB). Host-side attr; capture-safe.
  static_cast<void>(hipFuncSetAttribute(
      reinterpret_cast<const void*>(spec_stage1),
      hipFuncAttributeMaxDynamicSharedMemorySize, (int)sh1));
  static_cast<void>(hipFuncSetAttribute(
      reinterpret_cast<const void*>(spec_stage2),
      hipFuncAttributeMaxDynamicSharedMemorySize, (int)sh2));

  spec_stage1<<<dim3(NCHUNK, KG), 256, sh1, stream>>>(U, X, partials);
  spec_reduce<<<dim3((K_EIG * F_DIM) / 256), 256, 0, stream>>>(partials, g, specT);
  spec_stage2<<<dim3((N_NODES + 127) / 128), 256, sh2, stream>>>(U, specT, out);
}